// PointSetAttention_29832842838767
// MI455X (gfx1250) — compile-verified
//
#include <hip/hip_runtime.h>
#include <math.h>

typedef __attribute__((ext_vector_type(2))) float v2f;
typedef __attribute__((ext_vector_type(8))) float v8f;

#define C_CH   128
#define H_HEADS 4
#define M_TOT  384   // 3*C

// ---------------------------------------------------------------------------
// Kernel 0: init segmax=-inf, segsum=0, out=0
// ---------------------------------------------------------------------------
__global__ void psa_init_kernel(float* __restrict__ segmax,
                                float* __restrict__ segsum,
                                float* __restrict__ out,
                                int nH, long long nC) {
  long long i = (long long)blockIdx.x * blockDim.x + threadIdx.x;
  if (i < nH) { segmax[i] = -INFINITY; segsum[i] = 0.0f; }
  if (i < nC) out[i] = 0.0f;
}

// ---------------------------------------------------------------------------
// Kernel 1: QKV = feat @ qkv_w^T + qkv_b, using V_WMMA_F32_16X16X4_F32.
// One wave computes a 16(row) x 64(col) strip: 4 accumulators share one A
// fragment per k-step. N (=50000) and 384 are multiples of 16 -> EXEC all 1s.
//
// Fragment layouts (ISA 7.12.2, 32-bit, wave32):
//   A (16x4, MxK): lane L -> row M = L&15; v0 = K=2*(L>>4), v1 = K=2*(L>>4)+1
//   B (4x16, KxN): lane L -> col N = L&15; v0 = K=2*(L>>4), v1 = K=2*(L>>4)+1
//   D (16x16):     VGPR v: lanes 0-15 -> M=v, N=L; lanes 16-31 -> M=v+8, N=L-16
// ---------------------------------------------------------------------------
__global__ __launch_bounds__(32) void psa_qkv_gemm_kernel(
    const float* __restrict__ feat,   // N x 128
    const float* __restrict__ qkv_w,  // 384 x 128 (row-major)
    const float* __restrict__ qkv_b,  // 384
    float* __restrict__ Q,            // N x 128
    float* __restrict__ K,            // N x 128
    float* __restrict__ V,            // N x 128
    int N) {
  const int wid    = blockIdx.x;       // one wave per block
  const int mgroup = wid % (M_TOT / 64);
  const int ntile  = wid / (M_TOT / 64);
  const int n0     = ntile * 16;
  const int m0     = mgroup * 64;
  const int lane   = threadIdx.x;
  const int half   = lane >> 4;        // 0 or 1
  const int r      = lane & 15;

  v8f acc0 = {}, acc1 = {}, acc2 = {}, acc3 = {};
  const float* arow = feat + (size_t)(n0 + r) * C_CH + 2 * half;
  const float* b0   = qkv_w + (size_t)(m0 +  0 + r) * C_CH + 2 * half;
  const float* b1   = qkv_w + (size_t)(m0 + 16 + r) * C_CH + 2 * half;
  const float* b2   = qkv_w + (size_t)(m0 + 32 + r) * C_CH + 2 * half;
  const float* b3   = qkv_w + (size_t)(m0 + 48 + r) * C_CH + 2 * half;

  #pragma unroll 4
  for (int c0 = 0; c0 < C_CH; c0 += 4) {
    v2f a  = *(const v2f*)(arow + c0);
    v2f w0 = *(const v2f*)(b0 + c0);
    v2f w1 = *(const v2f*)(b1 + c0);
    v2f w2 = *(const v2f*)(b2 + c0);
    v2f w3 = *(const v2f*)(b3 + c0);
    acc0 = __builtin_amdgcn_wmma_f32_16x16x4_f32(false, a, false, w0, (short)0, acc0, false, false);
    acc1 = __builtin_amdgcn_wmma_f32_16x16x4_f32(false, a, false, w1, (short)0, acc1, false, false);
    acc2 = __builtin_amdgcn_wmma_f32_16x16x4_f32(false, a, false, w2, (short)0, acc2, false, false);
    acc3 = __builtin_amdgcn_wmma_f32_16x16x4_f32(false, a, false, w3, (short)0, acc3, false, false);
  }

  v8f accs[4] = {acc0, acc1, acc2, acc3};
  #pragma unroll
  for (int t = 0; t < 4; ++t) {
    int col = m0 + 16 * t + r;          // output channel in [0,384)
    float bias = qkv_b[col];
    float* outp; int cc;
    if (col < C_CH)          { outp = Q; cc = col; }
    else if (col < 2 * C_CH) { outp = K; cc = col - C_CH; }
    else                     { outp = V; cc = col - 2 * C_CH; }
    #pragma unroll
    for (int vi = 0; vi < 8; ++vi) {
      int row = n0 + vi + 8 * half;
      outp[(size_t)row * C_CH + cc] = accs[t][vi] + bias;
    }
  }
}

// ---------------------------------------------------------------------------
// float atomic max via int/uint bit trick (IEEE monotonicity)
// ---------------------------------------------------------------------------
__device__ __forceinline__ void atomicMaxF(float* addr, float val) {
  if (val >= 0.0f)
    atomicMax((int*)addr, __float_as_int(val));
  else
    atomicMin((unsigned int*)addr, (unsigned int)__float_as_int(val));
}

// ---------------------------------------------------------------------------
// Kernel 2: per-edge attention logits + segment max.
// One wave per edge; lane handles 4 contiguous channels; 8-lane groups = heads.
// ---------------------------------------------------------------------------
__global__ __launch_bounds__(256) void psa_edge_logits_kernel(
    const int*   __restrict__ graph,  // [2, E]
    const float* __restrict__ coord,  // N x 3
    const float* __restrict__ rpe_w,  // 128 x 3
    const float* __restrict__ rpe_b,  // 128
    const float* __restrict__ Q,
    const float* __restrict__ K,
    float* __restrict__ logits,       // E x H
    float* __restrict__ segmax,       // N x H
    int E) {
  int wave = (blockIdx.x * blockDim.x + threadIdx.x) >> 5;
  int lane = threadIdx.x & 31;
  if (wave >= E) return;

  int dst = graph[wave];
  int src = graph[E + wave];
  float rx = coord[(size_t)dst * 3 + 0] - coord[(size_t)src * 3 + 0];
  float ry = coord[(size_t)dst * 3 + 1] - coord[(size_t)src * 3 + 1];
  float rz = coord[(size_t)dst * 3 + 2] - coord[(size_t)src * 3 + 2];

  int c0 = lane * 4;
  float4 kv = *(const float4*)(K + (size_t)src * C_CH + c0);
  float4 qv = *(const float4*)(Q + (size_t)dst * C_CH + c0);

  float partial = 0.0f;
  const float* kk = (const float*)&kv;
  const float* qq = (const float*)&qv;
  #pragma unroll
  for (int i = 0; i < 4; ++i) {
    int c = c0 + i;
    float rpe = rpe_w[c * 3 + 0] * rx + rpe_w[c * 3 + 1] * ry +
                rpe_w[c * 3 + 2] * rz + rpe_b[c];
    partial += kk[i] * qq[i] + rpe;
  }
  // reduce across the 8 lanes of each head (lanes [8h, 8h+7])
  partial += __shfl_xor(partial, 1, 32);
  partial += __shfl_xor(partial, 2, 32);
  partial += __shfl_xor(partial, 4, 32);

  if ((lane & 7) == 0) {
    int h = lane >> 3;
    logits[(size_t)wave * H_HEADS + h] = partial;
    atomicMaxF(&segmax[(size_t)dst * H_HEADS + h], partial);
  }
}

// ---------------------------------------------------------------------------
// Kernel 3: ex = exp(logit - segmax[dst]); segsum[dst] += ex. One thread per
// (edge, head). Matches the reference's isfinite(seg_max) guard.
// ---------------------------------------------------------------------------
__global__ __launch_bounds__(256) void psa_edge_exp_kernel(
    const int*   __restrict__ graph,
    float*       __restrict__ logits,   // in: logit, out: ex
    const float* __restrict__ segmax,
    float*       __restrict__ segsum,
    int E) {
  int i = blockIdx.x * blockDim.x + threadIdx.x;
  if (i >= E * H_HEADS) return;
  int e = i >> 2;           // H_HEADS == 4
  int h = i & 3;
  int dst = graph[e];
  float m = segmax[(size_t)dst * H_HEADS + h];
  if (!isfinite(m)) m = 0.0f;
  float ex = expf(logits[i] - m);
  logits[i] = ex;
  atomicAdd(&segsum[(size_t)dst * H_HEADS + h], ex);
}

// ---------------------------------------------------------------------------
// Kernel 4: out[dst] += (ex / segsum[dst]) * V[src]. One wave per edge,
// lane covers 4 contiguous channels -> 4 f32 atomic adds per lane.
// ---------------------------------------------------------------------------
__global__ __launch_bounds__(256) void psa_scatter_kernel(
    const int*   __restrict__ graph,
    const float* __restrict__ V,
    const float* __restrict__ logits,   // holds ex
    const float* __restrict__ segsum,
    float* __restrict__ out,            // N x 128
    int E) {
  int wave = (blockIdx.x * blockDim.x + threadIdx.x) >> 5;
  int lane = threadIdx.x & 31;
  if (wave >= E) return;

  int dst = graph[wave];
  int src = graph[E + wave];
  int h = lane >> 3;                 // channels [32h, 32h+31]
  float w = logits[(size_t)wave * H_HEADS + h] /
            segsum[(size_t)dst * H_HEADS + h];

  int c0 = lane * 4;
  float4 vv = *(const float4*)(V + (size_t)src * C_CH + c0);
  float* op = out + (size_t)dst * C_CH + c0;
  atomicAdd(op + 0, w * vv.x);
  atomicAdd(op + 1, w * vv.y);
  atomicAdd(op + 2, w * vv.z);
  atomicAdd(op + 3, w * vv.w);
}

// ---------------------------------------------------------------------------
// launch
// ---------------------------------------------------------------------------
extern "C" void kernel_launch(void* const* d_in, const int* in_sizes, int n_in,
                              void* d_out, int out_size, void* d_ws, size_t ws_size,
                              hipStream_t stream) {
  const float* feat  = (const float*)d_in[0];
  const float* coord = (const float*)d_in[1];
  const int*   graph = (const int*)  d_in[2];
  const float* qkv_w = (const float*)d_in[3];
  const float* qkv_b = (const float*)d_in[4];
  const float* rpe_w = (const float*)d_in[5];
  const float* rpe_b = (const float*)d_in[6];
  float* out = (float*)d_out;

  const int N = in_sizes[0] / C_CH;   // 50000 (multiple of 16)
  const int E = in_sizes[2] / 2;      // 800000

  // workspace layout (floats)
  float* ws = (float*)d_ws;
  size_t NC = (size_t)N * C_CH;
  float* Q      = ws;
  float* K      = ws + NC;
  float* V      = ws + 2 * NC;
  float* logits = ws + 3 * NC;                       // E*H
  float* segmax = ws + 3 * NC + (size_t)E * H_HEADS; // N*H
  float* segsum = segmax + (size_t)N * H_HEADS;      // N*H

  // 0) init (covers max(N*H, N*C) = N*C elements)
  {
    long long total = (long long)NC;
    int blocks = (int)((total + 255) / 256);
    psa_init_kernel<<<blocks, 256, 0, stream>>>(segmax, segsum, out,
                                                N * H_HEADS, total);
  }
  // 1) QKV GEMM (WMMA f32 16x16x4): one wave per 16x64 strip
  {
    int blocks = (N / 16) * (M_TOT / 64);
    psa_qkv_gemm_kernel<<<blocks, 32, 0, stream>>>(feat, qkv_w, qkv_b,
                                                   Q, K, V, N);
  }
  // 2) edge logits + segment max (8 edges / 256-thread block)
  {
    int blocks = (E + 7) / 8;
    psa_edge_logits_kernel<<<blocks, 256, 0, stream>>>(
        graph, coord, rpe_w, rpe_b, Q, K, logits, segmax, E);
  }
  // 3) exp + segment sum
  {
    int blocks = (E * H_HEADS + 255) / 256;
    psa_edge_exp_kernel<<<blocks, 256, 0, stream>>>(graph, logits, segmax,
                                                    segsum, E);
  }
  // 4) weighted scatter
  {
    int blocks = (E + 7) / 8;
    psa_scatter_kernel<<<blocks, 256, 0, stream>>>(graph, V, logits, segsum,
                                                   out, E);
  }
}